// QuantumVariationalLayer_64759516889560
// MI455X (gfx1250) — compile-verified
//
#include <hip/hip_runtime.h>
#include <math.h>

typedef __attribute__((ext_vector_type(2))) float v2f;
typedef __attribute__((ext_vector_type(8))) float v8f;

#define NQ      12
#define NLAYERS 6
#define DIM     4096      // 2^NQ
#define D_IN    512
#define PI_F    3.14159265358979323846f

// ---------------------------------------------------------------------------
// Kernel 1: x = tanh(v @ W^T + b) * pi  via V_WMMA_F32_16X16X4_F32
//   M = B rows of v, N = 12 (padded to 16), K = 512.
//   One wave per 16x16 output tile; 8 waves (128 rows) per block.
// ---------------------------------------------------------------------------
__global__ __launch_bounds__(256) void angles_gemm_kernel(
    const float* __restrict__ v, const float* __restrict__ W,
    const float* __restrict__ bias, float* __restrict__ x, int B)
{
    // W (12x512) zero-padded to 16 rows; row stride 516 words avoids
    // bank conflicts on the per-lane row reads (4*lane mod 64 all distinct).
    __shared__ float Wlds[16][D_IN + 4];
    const int tid = threadIdx.x;
    for (int idx = tid; idx < 16 * D_IN; idx += 256) {
        int r = idx / D_IN, c = idx - r * D_IN;
        Wlds[r][c] = (r < NQ) ? W[r * D_IN + c] : 0.0f;
    }
    __syncthreads();

    const int wave = tid >> 5;
    const int lane = tid & 31;
    const int half = lane >> 4;    // 0: K={0,1}, 1: K={2,3}
    const int ln   = lane & 15;    // A row (M) / B col (N)
    const int m0   = blockIdx.x * 128 + wave * 16;
    const int kofs = half * 2;

    int m_a = m0 + ln;
    if (m_a >= B) m_a = B - 1;     // clamp (stores are guarded)
    const float* arow = v + (size_t)m_a * D_IN + kofs;

    v8f acc = {};
    for (int k = 0; k < D_IN; k += 4) {
        v2f a, b;
        a.x = arow[k];
        a.y = arow[k + 1];
        b.x = Wlds[ln][k + kofs];
        b.y = Wlds[ln][k + kofs + 1];
        acc = __builtin_amdgcn_wmma_f32_16x16x4_f32(
            /*neg_a=*/false, a, /*neg_b=*/false, b,
            /*c_mod=*/(short)0, acc, /*reuse_a=*/false, /*reuse_b=*/false);
    }

    // C/D layout: n = lane&15 ; m = vgpr_idx + 8*(lane>=16)
    const int n = ln;
    if (n < NQ) {
        const float bn = bias[n];
#pragma unroll
        for (int r = 0; r < 8; ++r) {
            const int m = m0 + r + half * 8;
            if (m < B) x[(size_t)m * NQ + n] = tanhf(acc[r] + bn) * PI_F;
        }
    }
}

// ---------------------------------------------------------------------------
// Kernel 2: 12-qubit state-vector simulation, one block per batch element.
//   State (4096 complex) lives entirely in LDS (32 KB re+im).
//   Qubit w corresponds to bit (NQ-1-w) of the flat amplitude index.
// ---------------------------------------------------------------------------
__global__ __launch_bounds__(256) void circuit_kernel(
    const float* __restrict__ x, const float* __restrict__ wts,
    float* __restrict__ out)
{
    __shared__ float sre[DIM];
    __shared__ float sim[DIM];
    __shared__ float gates[NLAYERS * NQ * 8]; // per-gate: u00 u01 u10 u11 (re,im)
    __shared__ float ry_c[NQ], ry_s[NQ];
    __shared__ float zacc[NQ];

    const int tid = threadIdx.x;
    const int b   = blockIdx.x;

    for (int i = tid; i < DIM; i += 256) { sre[i] = 0.0f; sim[i] = 0.0f; }
    if (tid == 0) sre[0] = 1.0f;
    if (tid < NQ) {
        const float th = x[(size_t)b * NQ + tid] * 0.5f;
        ry_c[tid] = cosf(th);
        ry_s[tid] = sinf(th);
        zacc[tid] = 0.0f;
    }
    // Rot(phi,theta,omega) = RZ(omega) RY(theta) RZ(phi)  (shared over batch)
    if (tid < NLAYERS * NQ) {
        const float* wp = wts + tid * 3;
        const float phi = wp[0], th = wp[1], om = wp[2];
        const float ch = cosf(th * 0.5f), sh = sinf(th * 0.5f);
        float ssp, csp, sdm, cdm;
        __sincosf((phi + om) * 0.5f, &ssp, &csp);
        __sincosf((phi - om) * 0.5f, &sdm, &cdm);
        float* g = &gates[tid * 8];
        g[0] =  csp * ch;  g[1] = -ssp * ch;   // u00 = e^{-i(phi+om)/2} c
        g[2] = -cdm * sh;  g[3] = -sdm * sh;   // u01 = -e^{+i(phi-om)/2} s
        g[4] =  cdm * sh;  g[5] = -sdm * sh;   // u10 =  e^{-i(phi-om)/2} s
        g[6] =  csp * ch;  g[7] =  ssp * ch;   // u11 = e^{+i(phi+om)/2} c
    }
    __syncthreads();

    // ---- AngleEmbedding: per-batch real RY gates ----
    for (int w = 0; w < NQ; ++w) {
        const int bit  = NQ - 1 - w;
        const int mask = 1 << bit;
        const float c = ry_c[w], s = ry_s[w];
        for (int p = tid; p < DIM / 2; p += 256) {
            const int i0 = ((p >> bit) << (bit + 1)) | (p & (mask - 1));
            const int i1 = i0 | mask;
            const float a0r = sre[i0], a0i = sim[i0];
            const float a1r = sre[i1], a1i = sim[i1];
            sre[i0] = c * a0r - s * a1r;  sim[i0] = c * a0i - s * a1i;
            sre[i1] = s * a0r + c * a1r;  sim[i1] = s * a0i + c * a1i;
        }
        __syncthreads();
    }

    // ---- StronglyEntanglingLayers ----
    for (int l = 0; l < NLAYERS; ++l) {
        for (int w = 0; w < NQ; ++w) {
            const float* g = &gates[(l * NQ + w) * 8];
            const float u00r = g[0], u00i = g[1], u01r = g[2], u01i = g[3];
            const float u10r = g[4], u10i = g[5], u11r = g[6], u11i = g[7];
            const int bit  = NQ - 1 - w;
            const int mask = 1 << bit;
            for (int p = tid; p < DIM / 2; p += 256) {
                const int i0 = ((p >> bit) << (bit + 1)) | (p & (mask - 1));
                const int i1 = i0 | mask;
                const float a0r = sre[i0], a0i = sim[i0];
                const float a1r = sre[i1], a1i = sim[i1];
                sre[i0] = u00r*a0r - u00i*a0i + u01r*a1r - u01i*a1i;
                sim[i0] = u00r*a0i + u00i*a0r + u01r*a1i + u01i*a1r;
                sre[i1] = u10r*a0r - u10i*a0i + u11r*a1r - u11i*a1i;
                sim[i1] = u10r*a0i + u10i*a0r + u11r*a1i + u11i*a1r;
            }
            __syncthreads();
        }
        const int r = (l % (NQ - 1)) + 1;
        for (int w = 0; w < NQ; ++w) {
            const int cmask = 1 << (NQ - 1 - w);
            const int tmask = 1 << (NQ - 1 - ((w + r) % NQ));
            for (int i = tid; i < DIM; i += 256) {
                if ((i & cmask) && !(i & tmask)) {
                    const int j = i | tmask;
                    const float tr = sre[i]; sre[i] = sre[j]; sre[j] = tr;
                    const float ti = sim[i]; sim[i] = sim[j]; sim[j] = ti;
                }
            }
            __syncthreads();
        }
    }

    // ---- PauliZ expectation values ----
    float acc[NQ];
#pragma unroll
    for (int w = 0; w < NQ; ++w) acc[w] = 0.0f;
    for (int i = tid; i < DIM; i += 256) {
        const float pr = sre[i], pim = sim[i];
        const float p = pr * pr + pim * pim;
#pragma unroll
        for (int w = 0; w < NQ; ++w)
            acc[w] += ((i >> (NQ - 1 - w)) & 1) ? -p : p;
    }
#pragma unroll
    for (int w = 0; w < NQ; ++w) {
#pragma unroll
        for (int off = 16; off >= 1; off >>= 1)
            acc[w] += __shfl_xor(acc[w], off, 32);
    }
    if ((tid & 31) == 0) {
#pragma unroll
        for (int w = 0; w < NQ; ++w) atomicAdd(&zacc[w], acc[w]);
    }
    __syncthreads();
    if (tid < NQ) out[(size_t)b * NQ + tid] = zacc[tid];
}

// ---------------------------------------------------------------------------
extern "C" void kernel_launch(void* const* d_in, const int* in_sizes, int n_in,
                              void* d_out, int out_size, void* d_ws, size_t ws_size,
                              hipStream_t stream)
{
    const float* v    = (const float*)d_in[0];   // (B, 512)
    const float* W    = (const float*)d_in[1];   // (12, 512)
    const float* bias = (const float*)d_in[2];   // (12,)
    const float* wts  = (const float*)d_in[3];   // (6, 12, 3)
    float* out = (float*)d_out;                  // (B, 12)

    const int B = in_sizes[0] / D_IN;            // 4096
    float* x = (float*)d_ws;                     // (B, 12) angle workspace

    const int tilesM = (B + 127) / 128;
    angles_gemm_kernel<<<dim3(tilesM), dim3(256), 0, stream>>>(v, W, bias, x, B);
    circuit_kernel<<<dim3(B), dim3(256), 0, stream>>>(x, wts, out);
}